// EncoderLayer_36352603193491
// MI455X (gfx1250) — compile-verified
//
#include <hip/hip_runtime.h>

#define IMG_H 128
#define IMG_W 128
#define TILE_H 32
#define PW (IMG_W + 2)      // 130 padded width
#define PH (TILE_H + 2)     // 34 padded rows
#define NTHREADS 256

typedef float v4f __attribute__((ext_vector_type(4)));
typedef __attribute__((address_space(1))) int as1_int;
typedef __attribute__((address_space(3))) int as3_int;

#if defined(__has_builtin)
#if __has_builtin(__builtin_amdgcn_global_load_async_to_lds_b32) && \
    __has_builtin(__builtin_amdgcn_s_wait_asynccnt)
#define HAVE_ASYNC_LDS 1
#endif
#endif
#ifndef HAVE_ASYNC_LDS
#define HAVE_ASYNC_LDS 0
#endif

__global__ __launch_bounds__(NTHREADS)
void lbp_glt_kernel(const float* __restrict__ images,
                    const float* __restrict__ lat,
                    float* __restrict__ out)
{
    __shared__ float tile[PH * PW];
    __shared__ float thrS[8];

    const int tid = threadIdx.x;
    const int b   = blockIdx.x >> 2;   // 4 tiles per image (128/32)
    const int t   = blockIdx.x & 3;
    const int r0  = t * TILE_H;

    // ---- thresholds: relu(lat)+1e-5, normalize, cumsum*2-1 (uniform, scalar path) ----
    if (tid == 0) {
        float p[8];
        float s = 0.f;
        #pragma unroll
        for (int i = 0; i < 8; ++i) {
            float v = lat[i];
            v = v > 0.f ? v : 0.f;
            v += 1e-5f;
            p[i] = v;
            s += v;
        }
        const float inv = 1.f / s;
        float c = 0.f;
        #pragma unroll
        for (int i = 0; i < 8; ++i) {
            c += p[i] * inv;
            thrS[i] = c * 2.f - 1.f;
        }
    }

    const float* img = images + (size_t)b * (IMG_H * IMG_W);

    // ---- stage zero-padded (TILE_H+2) x (W+2) tile into LDS ----
    for (int idx = tid; idx < PH * PW; idx += NTHREADS) {
        const int pr = idx / PW;
        const int pc = idx - pr * PW;
        const int gr = r0 + pr - 1;
        const int gc = pc - 1;
        const bool inb = (gr >= 0) & (gr < IMG_H) & (gc >= 0) & (gc < IMG_W);
#if HAVE_ASYNC_LDS
        if (inb) {
            __builtin_amdgcn_global_load_async_to_lds_b32(
                (as1_int*)(img + gr * IMG_W + gc),
                (as3_int*)&tile[idx], 0, 0);
        } else {
            tile[idx] = 0.f;
        }
#else
        tile[idx] = inb ? img[gr * IMG_W + gc] : 0.f;
#endif
    }
#if HAVE_ASYNC_LDS
    __builtin_amdgcn_s_wait_asynccnt(0);
#endif
    __syncthreads();

    const float thr0 = thrS[0], thr1 = thrS[1], thr2 = thrS[2], thr3 = thrS[3];
    const float thr4 = thrS[4], thr5 = thrS[5], thr6 = thrS[6], thr7 = thrS[7];

    // ---- compute: 16 pixels per thread, two 128-bit NT stores each ----
    for (int p = tid; p < TILE_H * IMG_W; p += NTHREADS) {
        const int lr = p >> 7;        // p / 128
        const int lc = p & 127;       // p % 128
        const float* c0 = &tile[(lr + 1) * PW + (lc + 1)];
        const float ctr = c0[0];

        float code = 0.f;
        code += (c0[-PW - 1] >= ctr) ?   1.f : 0.f;  // (-1,-1) bit0
        code += (c0[-PW    ] >= ctr) ?   2.f : 0.f;  // (-1, 0) bit1
        code += (c0[-PW + 1] >= ctr) ?   4.f : 0.f;  // (-1, 1) bit2
        code += (c0[      1] >= ctr) ?   8.f : 0.f;  // ( 0, 1) bit3
        code += (c0[ PW + 1] >= ctr) ?  16.f : 0.f;  // ( 1, 1) bit4
        code += (c0[ PW    ] >= ctr) ?  32.f : 0.f;  // ( 1, 0) bit5
        code += (c0[ PW - 1] >= ctr) ?  64.f : 0.f;  // ( 1,-1) bit6
        code += (c0[     -1] >= ctr) ? 128.f : 0.f;  // ( 0,-1) bit7

        const float x = code * (1.0f / 127.5f) - 1.0f;

        v4f o0, o1;
        o0.x = (x >= thr0) ? 1.f : 0.f;
        o0.y = (x >= thr1) ? 1.f : 0.f;
        o0.z = (x >= thr2) ? 1.f : 0.f;
        o0.w = (x >= thr3) ? 1.f : 0.f;
        o1.x = (x >= thr4) ? 1.f : 0.f;
        o1.y = (x >= thr5) ? 1.f : 0.f;
        o1.z = (x >= thr6) ? 1.f : 0.f;
        o1.w = (x >= thr7) ? 1.f : 0.f;

        const size_t base = (((size_t)(b * IMG_H + r0 + lr)) * IMG_W + lc) * 8;
        v4f* op = (v4f*)(out + base);
        __builtin_nontemporal_store(o0, op);
        __builtin_nontemporal_store(o1, op + 1);
    }
}

extern "C" void kernel_launch(void* const* d_in, const int* in_sizes, int n_in,
                              void* d_out, int out_size, void* d_ws, size_t ws_size,
                              hipStream_t stream) {
    const float* images = (const float*)d_in[0];
    const float* lat    = (const float*)d_in[1];
    float* out          = (float*)d_out;

    const int B = in_sizes[0] / (IMG_H * IMG_W);     // 256
    const int tilesPerImg = IMG_H / TILE_H;          // 4
    dim3 grid(B * tilesPerImg);
    dim3 block(NTHREADS);
    lbp_glt_kernel<<<grid, block, 0, stream>>>(images, lat, out);
}